// WavUnPacking_30090540876444
// MI455X (gfx1250) — compile-verified
//
#include <hip/hip_runtime.h>

// Inverse Haar (db1) DWT, shapes fixed by the reference:
//   in  x : (8, 256, 128, 128) f32   [ll | lh | hl | hh] along channels
//   out   : (8,  64, 256, 256) f32
//
// Strategy: per-position 4-vector butterfly executed on the CDNA5 matrix unit
// via V_WMMA_F32_16X16X4_F32:  D(16x16) = A(16x4 data) x B(4x16 = 0.5*H^T
// replicated).  A and D VGPR layouts are the ISA-documented f32 layouts, so
// loads are two fully-utilized 2x64B global_load_b32 per wave and stores
// cover two contiguous 128B output-row segments per wave.

typedef __attribute__((ext_vector_type(2))) float v2f;
typedef __attribute__((ext_vector_type(8))) float v8f;

#define BATCH 8
#define C4    64   // output channels
#define H     128
#define W     128

__global__ __launch_bounds__(256) void haar_idwt_wmma(const float* __restrict__ x,
                                                      float* __restrict__ out) {
    const int lane = threadIdx.x & 31;
    const int waveInBlock = threadIdx.x >> 5;
    const unsigned wid = blockIdx.x * 8u + waveInBlock;   // global wave id

    // wid = ((b*C4 + c)*H + i)*8 + q ; each wave: 16 cols starting at j0=16q
    const int q = wid & 7;
    unsigned t = wid >> 3;
    const int i = t & (H - 1);
    t >>= 7;
    const int c = t & (C4 - 1);
    const int b = t >> 6;
    const int j0 = q * 16;

    const int  m  = lane & 15;      // position within the 16-col chunk
    const bool hi = lane >= 16;

    // ---- A matrix (data), documented 16x4 f32 layout ----
    // lanes 0-15 : a.x = K0 = ll[pos], a.y = K1 = lh[pos]
    // lanes 16-31: a.x = K2 = hl[pos], a.y = K3 = hh[pos]
    const size_t plane = (size_t)H * W;
    const size_t base0 = ((size_t)b * (4 * C4) + c) * plane + (size_t)i * W + j0 + m;
    const size_t g0 = hi ? 2u : 0u;   // channel-group for a.x
    const size_t g1 = hi ? 3u : 1u;   // channel-group for a.y
    v2f a;
    a.x = x[base0 + g0 * (C4 * plane)];
    a.y = x[base0 + g1 * (C4 * plane)];

    // ---- B matrix (constants): B[k][n] = 0.5 * Hrow[n%4][k] ----
    // Haar synthesis rows over [ll,lh,hl,hh]:
    //   e00:[+ + + +]  e01:[+ - + -]  e10:[+ + - -]  e11:[+ - - +]
    // Layout: rows striped across lanes; VGPR0 = {K0: lanes0-15, K2: lanes16-31},
    //         VGPR1 = {K1: lanes0-15, K3: lanes16-31}
    const int s = lane & 3;                             // component class n%4
    const float h_k0 = 1.0f;
    const float h_k1 = (s < 2) ? 1.0f : -1.0f;
    const float h_k2 = (s & 1) ? -1.0f : 1.0f;
    const float h_k3 = (s == 0 || s == 3) ? 1.0f : -1.0f;
    v2f bm;
    bm.x = 0.5f * (hi ? h_k1 : h_k0);
    bm.y = 0.5f * (hi ? h_k3 : h_k2);

    // ---- matrix-unit butterfly (EXEC is all-ones here: uniform flow) ----
    v8f cz = {};
    v8f d = __builtin_amdgcn_wmma_f32_16x16x4_f32(
        /*neg_a=*/false, a, /*neg_b=*/false, bm,
        /*c_mod=*/(short)0, cz, /*reuse_a=*/false, /*reuse_b=*/false);

    // ---- store: D VGPR r, lane n: e_{n%4}[pos j0 + r + (hi?8:0)] ----
    // Lanes with (lane&15) < 4 carry the 4 distinct components (others dup).
    if ((lane & 15) < 4) {
        const int sc = lane & 3;       // component 0..3
        const int di = sc >> 1;        // sub-row  (0 -> 2i, 1 -> 2i+1)
        const int dj = sc & 1;         // sub-col
        const size_t orow =
            ((size_t)(b * C4 + c) * (2 * H) + (size_t)(2 * i + di)) * (2 * W);
        const int pbase = j0 + (hi ? 8 : 0);
#pragma unroll
        for (int r = 0; r < 8; ++r) {
            out[orow + 2 * (pbase + r) + dj] = d[r];
        }
    }
}

extern "C" void kernel_launch(void* const* d_in, const int* in_sizes, int n_in,
                              void* d_out, int out_size, void* d_ws, size_t ws_size,
                              hipStream_t stream) {
    const float* x = (const float*)d_in[0];
    float* out = (float*)d_out;

    // 64 input floats (16 positions x 4 channels) per wave, 8 waves per block
    const int total  = in_sizes[0];          // 33,554,432
    const int waves  = total / 64;           // 524,288
    const int blocks = waves / 8;            // 65,536
    haar_idwt_wmma<<<blocks, 256, 0, stream>>>(x, out);
}